// GCN_20590073217488
// MI455X (gfx1250) — compile-verified
//
#include <hip/hip_runtime.h>

#define D 128

typedef __attribute__((ext_vector_type(2))) float v2f;
typedef __attribute__((ext_vector_type(8))) float v8f;

// ---------------------------------------------------------------------------
// Degree / normalization helpers
// ---------------------------------------------------------------------------
__global__ __launch_bounds__(256) void k_init_deg(float* deg, int N) {
    int i = blockIdx.x * 256 + threadIdx.x;
    if (i < N) deg[i] = 1.0f;  // self-loop contributes 1 to every node's degree
}

__global__ __launch_bounds__(256) void k_count_deg(const int* __restrict__ dst,
                                                   float* deg, int E) {
    int i = blockIdx.x * 256 + threadIdx.x;
    if (i < E) atomicAdd(&deg[dst[i]], 1.0f);
}

__global__ __launch_bounds__(256) void k_rsqrt(float* deg, int N) {
    int i = blockIdx.x * 256 + threadIdx.x;
    if (i < N) deg[i] = rsqrtf(deg[i]);  // in-place: deg -> dinv
}

__global__ __launch_bounds__(256) void k_zero(float* p, size_t n) {
    size_t i = (size_t)blockIdx.x * 256 + threadIdx.x;
    if (i < n) p[i] = 0.0f;
}

// ---------------------------------------------------------------------------
// Dense transform: C[M x 128] = A[M x 128] * W[128 x 128], full f32 via
// V_WMMA_F32_16X16X4_F32. One wave owns a 16x16 C tile; a 256-thread block
// (8 waves) covers 16 rows x 128 cols. Requires M % 16 == 0 (50000 = 3125*16).
//
// 32-bit A-matrix 16x4 lane layout (ISA 7.12.2):
//   lanes 0-15 : row = lane,    v0 = K+0, v1 = K+1
//   lanes 16-31: row = lane-16, v0 = K+2, v1 = K+3
// B (4x16, K-major) is the mirrored layout with "row" -> "col".
// C/D 16x16 f32: VGPR r -> (M = r + 8*(lane>=16), N = lane&15) within tile.
// ---------------------------------------------------------------------------
__global__ __launch_bounds__(256) void k_gemm_wmma(const float* __restrict__ A,
                                                   const float* __restrict__ W,
                                                   float* __restrict__ C) {
    const int wave  = threadIdx.x >> 5;
    const int lane  = threadIdx.x & 31;
    const int row0  = blockIdx.x * 16;
    const int col0  = wave * 16;            // N = 128 -> 8 column tiles
    const int lrow  = lane & 15;            // A-row / B-col within tile
    const int khalf = (lane >> 4) * 2;      // 0 for lanes 0-15, 2 for 16-31

    const float* Arow = A + (size_t)(row0 + lrow) * D;

    v8f acc = {};
    for (int k = 0; k < D; k += 4) {
        v2f a;
        a.x = Arow[k + khalf + 0];
        a.y = Arow[k + khalf + 1];
        v2f b;
        b.x = W[(size_t)(k + khalf + 0) * D + col0 + lrow];
        b.y = W[(size_t)(k + khalf + 1) * D + col0 + lrow];
        acc = __builtin_amdgcn_wmma_f32_16x16x4_f32(
            /*neg_a=*/false, a, /*neg_b=*/false, b,
            /*c_mod=*/(short)0, acc, /*reuse_a=*/false, /*reuse_b=*/false);
    }

    const int m_add = (lane >> 4) * 8;
    const int ncol  = col0 + lrow;
#pragma unroll
    for (int r = 0; r < 8; ++r) {
        C[(size_t)(row0 + r + m_add) * D + ncol] = acc[r];
    }
}

// ---------------------------------------------------------------------------
// Edge aggregation: one wave32 per edge; each lane moves a float4 (4 columns).
// agg[dst] += h[src] * dinv[src] * dinv[dst]
// ---------------------------------------------------------------------------
__global__ __launch_bounds__(256) void k_scatter(const float* __restrict__ h,
                                                 const int* __restrict__ src,
                                                 const int* __restrict__ dst,
                                                 const float* __restrict__ dinv,
                                                 float* agg, int E) {
    int e    = (int)(((size_t)blockIdx.x * 256 + threadIdx.x) >> 5);
    int lane = threadIdx.x & 31;
    if (e >= E) return;

    int s = src[e];
    int d = dst[e];
    float nrm = dinv[s] * dinv[d];

    const float4* hs = (const float4*)(h + (size_t)s * D);
    float4 v = hs[lane];                       // coalesced 512B per wave

    float* a = agg + (size_t)d * D + lane * 4;
    atomicAdd(a + 0, v.x * nrm);
    atomicAdd(a + 1, v.y * nrm);
    atomicAdd(a + 2, v.z * nrm);
    atomicAdd(a + 3, v.w * nrm);
}

// ---------------------------------------------------------------------------
// Finalize: out = (agg + h * dinv^2  [self-loop] + bias), optional ReLU.
// Safe to run in-place (out == agg).
// ---------------------------------------------------------------------------
__global__ __launch_bounds__(256) void k_finalize(const float* __restrict__ agg,
                                                  const float* __restrict__ h,
                                                  const float* __restrict__ dinv,
                                                  const float* __restrict__ bias,
                                                  float* __restrict__ out,
                                                  size_t ND, int do_relu) {
    size_t i = (size_t)blockIdx.x * 256 + threadIdx.x;
    if (i >= ND) return;
    int n = (int)(i >> 7);    // / 128
    int c = (int)(i & 127);
    float di = dinv[n];
    float v = agg[i] + h[i] * di * di + bias[c];
    if (do_relu) v = fmaxf(v, 0.0f);
    out[i] = v;
}

// ---------------------------------------------------------------------------
// Orchestration
// ---------------------------------------------------------------------------
extern "C" void kernel_launch(void* const* d_in, const int* in_sizes, int n_in,
                              void* d_out, int out_size, void* d_ws, size_t ws_size,
                              hipStream_t stream) {
    const float* x  = (const float*)d_in[0];
    const int*   ei = (const int*)d_in[1];
    const float* W1 = (const float*)d_in[2];
    const float* b1 = (const float*)d_in[3];
    const float* W2 = (const float*)d_in[4];
    const float* b2 = (const float*)d_in[5];
    float* out = (float*)d_out;

    const int N = in_sizes[0] / D;      // 50000
    const int E = in_sizes[1] / 2;      // 800000
    const int* src = ei;                // edge_index[0]
    const int* dst = ei + E;            // edge_index[1]
    const size_t ND = (size_t)N * D;

    // Workspace layout (floats): dinv[N] | h[N*D] | agg[N*D]   (~51.6 MB)
    float* ws   = (float*)d_ws;
    float* dinv = ws;
    float* h    = ws + 65536;           // 256 KB offset, 16B-aligned for float4
    float* agg  = h + ND;

    const int nb_N  = (N + 255) / 256;
    const int nb_E  = (E + 255) / 256;
    const int nb_ED = (E + 7) / 8;                  // 1 wave per edge, 8 waves/block
    const int nb_ND = (int)((ND + 255) / 256);
    const int nb_M  = N / 16;                       // 3125 row tiles (exact)

    // --- degree / normalization ---
    k_init_deg <<<nb_N, 256, 0, stream>>>(dinv, N);
    k_count_deg<<<nb_E, 256, 0, stream>>>(dst, dinv, E);
    k_rsqrt    <<<nb_N, 256, 0, stream>>>(dinv, N);

    // --- layer 1: h = x @ W1 ; agg = scatter(h) ; agg = relu(agg + self + b1)
    k_gemm_wmma<<<nb_M, 256, 0, stream>>>(x, W1, h);
    k_zero     <<<nb_ND, 256, 0, stream>>>(agg, ND);
    k_scatter  <<<nb_ED, 256, 0, stream>>>(h, src, dst, dinv, agg, E);
    k_finalize <<<nb_ND, 256, 0, stream>>>(agg, h, dinv, b1, agg, ND, 1);

    // --- layer 2: h = agg @ W2 ; out = scatter(h) ; out = out + self + b2
    k_gemm_wmma<<<nb_M, 256, 0, stream>>>(agg, W2, h);
    k_zero     <<<nb_ND, 256, 0, stream>>>(out, ND);
    k_scatter  <<<nb_ED, 256, 0, stream>>>(h, src, dst, dinv, out, E);
    k_finalize <<<nb_ND, 256, 0, stream>>>(out, h, dinv, b2, out, ND, 0);
}